// HashEncoder_2018634629279
// MI455X (gfx1250) — compile-verified
//
#include <hip/hip_runtime.h>

// Instant-NGP style multiresolution hash-grid encoder for MI455X (gfx1250).
// Memory-bound (~153 MB HBM traffic -> ~6.6us at 23.3 TB/s). Tables staged
// into LDS via gfx1250 async global->LDS copies; gathers served by ds_load_b32.

#define NLEVELS 16
#define TBL     16384
#define TMASK   (TBL - 1)
#define LPG     4                    // levels per y-group
#define NGROUPS (NLEVELS / LPG)      // 4
#define THREADS 512                  // 16 wave32 per WGP
#define PPT     8                    // points per thread
#define PPB     (THREADS * PPT)      // 4096 points per block
#define LDS_FLOATS (LPG * TBL)       // 65536 floats
#define LDS_BYTES  (LDS_FLOATS * 4)  // 262144 B (fits 320KB WGP LDS)

#define PRIME1 2654435761u
#define PRIME2 805459861u

// Pointee type the async-copy builtin expects: vector of 4 ints (16 bytes).
typedef int v4i __attribute__((ext_vector_type(4)));
typedef __attribute__((address_space(1))) v4i* as1_v4i_ptr;   // global
typedef __attribute__((address_space(3))) v4i* as3_v4i_ptr;   // LDS

// N_l = floor(16 * 2^(l/3)), l = 0..15
__constant__ float c_Nl[NLEVELS] = {
    16.f, 20.f, 25.f, 32.f, 40.f, 50.f, 64.f, 80.f,
    101.f, 128.f, 161.f, 203.f, 256.f, 322.f, 406.f, 512.f
};

__device__ __forceinline__ float lerp1(float a, float b, float w) {
    // a*(1-w) + b*w  (matches reference formulation)
    return fmaf(b, w, a * (1.0f - w));
}

__global__ __launch_bounds__(THREADS)
void hashgrid_encode_kernel(const float* __restrict__ x,
                            const float* __restrict__ tables,
                            float* __restrict__ out,
                            int npts) {
    extern __shared__ float smem[];          // LPG * TBL floats (256 KiB)
    const int group = blockIdx.y;            // which 4-level group
    const int l0    = group * LPG;
    const int tid   = threadIdx.x;

    // ---- Stage 4 level tables (256 KiB) global -> LDS -------------------
    const float* gsrc = tables + (size_t)l0 * TBL;   // tables: (L, T, 1) contiguous
#if defined(__gfx1250__) && __has_builtin(__builtin_amdgcn_global_load_async_to_lds_b128) && __has_builtin(__builtin_amdgcn_s_wait_asynccnt)
    for (int i = tid; i < LDS_FLOATS / 4; i += THREADS) {
        __builtin_amdgcn_global_load_async_to_lds_b128(
            (as1_v4i_ptr)(gsrc + 4 * i),
            (as3_v4i_ptr)(smem + 4 * i),
            /*imm offset*/0, /*cpol*/0);
    }
    __builtin_amdgcn_s_wait_asynccnt(0);
#else
    for (int i = tid; i < LDS_FLOATS / 4; i += THREADS) {
        const float4 v = reinterpret_cast<const float4*>(gsrc)[i];
        reinterpret_cast<float4*>(smem)[i] = v;
    }
#endif
    __syncthreads();

    // ---- Encode points ---------------------------------------------------
    const long long base = (long long)blockIdx.x * PPB;
    for (int it = 0; it < PPT; ++it) {
        const long long p = base + (long long)it * THREADS + tid;
        if (p >= npts) break;

#if defined(__gfx1250__)
        // speculative prefetch of next point tile -> global_prefetch_b8
        __builtin_prefetch(&x[3 * (p + THREADS)], 0, 1);
#endif
        const float px = x[3 * p + 0];
        const float py = x[3 * p + 1];
        const float pz = x[3 * p + 2];

        float r[LPG];
#pragma unroll
        for (int j = 0; j < LPG; ++j) {
            const float Nl = c_Nl[l0 + j];
            const float fx = px * Nl, fy = py * Nl, fz = pz * Nl;
            const float bx = floorf(fx), by = floorf(fy), bz = floorf(fz);
            const float wx = fx - bx, wy = fy - by, wz = fz - bz;

            const unsigned ux = (unsigned)(int)bx;
            const unsigned uy = (unsigned)(int)by;
            const unsigned uz = (unsigned)(int)bz;
            const unsigned hx0 = ux;                 // * PRIMES[0] == 1
            const unsigned hx1 = ux + 1u;
            const unsigned hy0 = uy * PRIME1;
            const unsigned hy1 = (uy + 1u) * PRIME1;
            const unsigned hz0 = uz * PRIME2;
            const unsigned hz1 = (uz + 1u) * PRIME2;

            const float* tb = smem + j * TBL;        // LDS gathers (ds_load_b32)
            const float v000 = tb[(hx0 ^ hy0 ^ hz0) & TMASK];
            const float v001 = tb[(hx0 ^ hy0 ^ hz1) & TMASK];
            const float v010 = tb[(hx0 ^ hy1 ^ hz0) & TMASK];
            const float v011 = tb[(hx0 ^ hy1 ^ hz1) & TMASK];
            const float v100 = tb[(hx1 ^ hy0 ^ hz0) & TMASK];
            const float v101 = tb[(hx1 ^ hy0 ^ hz1) & TMASK];
            const float v110 = tb[(hx1 ^ hy1 ^ hz0) & TMASK];
            const float v111 = tb[(hx1 ^ hy1 ^ hz1) & TMASK];

            const float x0 = lerp1(v000, v100, wx);  // lerp along dim0
            const float x1 = lerp1(v001, v101, wx);
            const float x2 = lerp1(v010, v110, wx);
            const float x3 = lerp1(v011, v111, wx);
            const float y0 = lerp1(x0, x2, wy);      // dim1
            const float y1 = lerp1(x1, x3, wy);
            r[j] = lerp1(y0, y1, wz);                // dim2
        }

        // out layout (B,N,16): levels l0..l0+3 are contiguous -> one b128 store
        float4 res;
        res.x = r[0]; res.y = r[1]; res.z = r[2]; res.w = r[3];
        reinterpret_cast<float4*>(out)[p * NGROUPS + group] = res;
    }
}

extern "C" void kernel_launch(void* const* d_in, const int* in_sizes, int n_in,
                              void* d_out, int out_size, void* d_ws, size_t ws_size,
                              hipStream_t stream) {
    const float* x      = (const float*)d_in[0];   // (8, 262144, 3) f32
    const float* tables = (const float*)d_in[1];   // (16, 16384, 1) f32
    float* out          = (float*)d_out;           // (8, 262144, 16) f32

    const long long npts = (long long)in_sizes[0] / 3;
    const int blocks_x = (int)((npts + PPB - 1) / PPB);   // 512 for 2,097,152 pts
    dim3 grid(blocks_x, NGROUPS, 1);
    dim3 block(THREADS, 1, 1);

    hipLaunchKernelGGL(hashgrid_encode_kernel, grid, block, LDS_BYTES, stream,
                       x, tables, out, (int)npts);
}